// NCL_71949292143322
// MI455X (gfx1250) — compile-verified
//
#include <hip/hip_runtime.h>
#include <stdint.h>

// LightGCN propagation for MI455X (gfx1250).
// Memory-bound (0.23 flop/byte): WMMA provably unhelpful (no dense tiles, and
// duplicate-dst merging saves ~0 atomics for random dst over 150k nodes).
// Working set (~155MB) fits the 192MB L2 -> gathers/atomics are L2-resident.
// CDNA5 paths used: global_load_async_to_lds_b128 (ASYNCcnt) double-buffering
// of the 256B/edge row gathers, with s_wait_asynccnt split-counter waits, plus
// pipelined edge-metadata loads so the post-wait path is pure LDS->VALU->atomic.

#define EMB 64
#define WAVES_PER_BLOCK 8
#define SCATTER_ITERS 8
#define EDGES_PER_BLOCK (WAVES_PER_BLOCK * 2 * SCATTER_ITERS)

// ---------------- copy: b[i] = a[i] = s[i] (float4) --------------------------
__global__ void __launch_bounds__(256) k_copy2(const float4* __restrict__ s,
                                               float4* __restrict__ a,
                                               float4* __restrict__ b,
                                               long long n4) {
  long long i = (long long)blockIdx.x * blockDim.x + threadIdx.x;
  if (i >= n4) return;
  float4 v = s[i];
  a[i] = v;
  b[i] = v;
}

// ---------------- degree: deg[dst] += 1 --------------------------------------
__global__ void __launch_bounds__(256) k_deg(const long long* __restrict__ dst,
                                             float* __restrict__ deg, int M) {
  int e = blockIdx.x * blockDim.x + threadIdx.x;
  if (e < M) atomicAdd(deg + (int)dst[e], 1.0f);
}

// ---------------- dinv: deg <- deg>0 ? rsqrt(deg) : 0 (in place) -------------
__global__ void __launch_bounds__(256) k_dinv(float* __restrict__ deg, int N) {
  int i = blockIdx.x * blockDim.x + threadIdx.x;
  if (i < N) {
    float d = deg[i];
    deg[i] = (d > 0.0f) ? rsqrtf(d) : 0.0f;
  }
}

// ---------------- edge scatter with async-LDS double buffering ---------------
// Each wave32 handles 2 edges per iteration: 16 lanes per edge, one float4
// chunk per lane. While iteration k's rows are consumed (LDS read, scale,
// 4x no-return global_atomic_add_f32), iteration k+1's rows are fetched with
// global_load_async_to_lds_b128 AND its metadata (dst, norm) is loaded, so the
// dependent src->dinv chain overlaps the async wait.
__global__ void __launch_bounds__(256) k_scatter(const float* __restrict__ x,
                                                 float* __restrict__ y,
                                                 const float* __restrict__ dinv,
                                                 const long long* __restrict__ src,
                                                 const long long* __restrict__ dst,
                                                 int M) {
  __shared__ float4 sbuf[WAVES_PER_BLOCK][2][2][16]; // [wave][buf][edge][chunk] = 8KB
  const int wave = threadIdx.x >> 5;
  const int lane = threadIdx.x & 31;
  const int ew   = lane >> 4;  // edge slot within wave (0/1)
  const int ch   = lane & 15;  // float4 chunk within 64-float row

  const int wbase = (blockIdx.x * WAVES_PER_BLOCK + wave) * (2 * SCATTER_ITERS) + ew;

  const uint32_t lds0 = (uint32_t)(size_t)(&sbuf[wave][0][ew][ch]);
  const uint32_t lds1 = (uint32_t)(size_t)(&sbuf[wave][1][ew][ch]);

  // Issue async row fetch for iteration k into the given LDS buffer.
  auto issue = [&](int k, uint32_t ldsoff) {
    int e = wbase + 2 * k;
    if (e >= M) e = M - 1;              // keep EXEC uniform; tail guarded at use
    int s = (int)src[e];
    const float* gp = x + (size_t)s * EMB + ch * 4;
    asm volatile("global_load_async_to_lds_b128 %0, %1, off"
                 :: "v"(ldsoff), "v"(gp)
                 : "memory");
  };
  // Fetch metadata (dst index + edge norm) for iteration k.
  auto meta = [&](int k, int& d, float& nrm, bool& ok) {
    int e = wbase + 2 * k;
    ok = (e < M);
    int ec = ok ? e : (M - 1);
    int s = (int)src[ec];
    d = (int)dst[ec];
    nrm = dinv[s] * dinv[d];
  };

  int   curD;  float curN;  bool curOk;
  int   nxtD;  float nxtN;  bool nxtOk;

  issue(0, lds0);
  meta(0, curD, curN, curOk);

#pragma unroll
  for (int k = 0; k < SCATTER_ITERS; ++k) {
    if (k + 1 < SCATTER_ITERS) {
      issue(k + 1, ((k & 1) ? lds0 : lds1));
      meta(k + 1, nxtD, nxtN, nxtOk);
      // async loads complete in order: <=1 outstanding => iter k's data landed
      asm volatile("s_wait_asynccnt 0x1" ::: "memory");
    } else {
      asm volatile("s_wait_asynccnt 0x0" ::: "memory");
    }
    if (curOk) {
      float4 v = sbuf[wave][k & 1][ew][ch];
      float* yp = y + (size_t)curD * EMB + ch * 4;
      atomicAdd(yp + 0, v.x * curN);
      atomicAdd(yp + 1, v.y * curN);
      atomicAdd(yp + 2, v.z * curN);
      atomicAdd(yp + 3, v.w * curN);
    }
    curD = nxtD; curN = nxtN; curOk = nxtOk;
  }
}

// ---------------- acc: out = (out + xn) * scale ------------------------------
__global__ void __launch_bounds__(256) k_acc(float4* __restrict__ out,
                                             const float4* __restrict__ xn,
                                             float scale, long long n4) {
  long long i = (long long)blockIdx.x * blockDim.x + threadIdx.x;
  if (i >= n4) return;
  float4 a = out[i];
  float4 b = xn[i];
  a.x = (a.x + b.x) * scale;
  a.y = (a.y + b.y) * scale;
  a.z = (a.z + b.z) * scale;
  a.w = (a.w + b.w) * scale;
  out[i] = a;
}

extern "C" void kernel_launch(void* const* d_in, const int* in_sizes, int n_in,
                              void* d_out, int out_size, void* d_ws, size_t ws_size,
                              hipStream_t stream) {
  const float*     ue = (const float*)d_in[0];
  const float*     ie = (const float*)d_in[1];
  const long long* ei = (const long long*)d_in[2];  // int64 edge_index [2, M]

  const long long U = in_sizes[0] / EMB;
  const long long I = in_sizes[1] / EMB;
  const long long N = U + I;
  const int       M = in_sizes[2] / 2;

  const long long* src = ei;
  const long long* dst = ei + M;

  float* xa  = (float*)d_ws;           // [N*64]
  float* xb  = xa + (size_t)N * EMB;   // [N*64]
  float* deg = xb + (size_t)N * EMB;   // [N] degrees, then dinv in place

  const long long nTot4 = N * (EMB / 4);
  const long long nU4   = U * (EMB / 4);
  const long long nI4   = I * (EMB / 4);

  // 1) x0 = concat(user, item); acc(d_out) = x0
  k_copy2<<<(int)((nU4 + 255) / 256), 256, 0, stream>>>(
      (const float4*)ue, (float4*)xa, (float4*)d_out, nU4);
  k_copy2<<<(int)((nI4 + 255) / 256), 256, 0, stream>>>(
      (const float4*)ie, (float4*)xa + nU4, (float4*)d_out + nU4, nI4);

  // 2) degrees -> dinv (in place)
  hipMemsetAsync(deg, 0, (size_t)N * sizeof(float), stream);
  k_deg<<<(M + 255) / 256, 256, 0, stream>>>(dst, deg, M);
  k_dinv<<<(int)((N + 255) / 256), 256, 0, stream>>>(deg, (int)N);

  // 3) three propagation layers, accumulate into d_out; fold /4 into last pass
  float* xc = xa;
  float* xn = xb;
  const int sblocks = (M + EDGES_PER_BLOCK - 1) / EDGES_PER_BLOCK;
  const int ablocks = (int)((nTot4 + 255) / 256);
  for (int l = 0; l < 3; ++l) {
    hipMemsetAsync(xn, 0, (size_t)N * EMB * sizeof(float), stream);
    k_scatter<<<sblocks, 256, 0, stream>>>(xc, xn, deg, src, dst, M);
    k_acc<<<ablocks, 256, 0, stream>>>((float4*)d_out, (const float4*)xn,
                                       (l == 2) ? 0.25f : 1.0f, nTot4);
    float* t = xc; xc = xn; xn = t;
  }
}